// GaussianSmoothing3d_721554506213
// MI455X (gfx1250) — compile-verified
//
#include <hip/hip_runtime.h>
#include <hip/hip_bf16.h>

typedef float v2f __attribute__((ext_vector_type(2)));
typedef float v8f __attribute__((ext_vector_type(8)));

#define DIMSZ 256
#define TS 16      // output tile edge
#define HS 20      // haloed tile edge (TS + 2*radius)
#define NTAP 5

// Register-only band select: returns g[d] for d in [0,5), else 0.
__device__ __forceinline__ float pick(int d, float g0, float g1, float g2,
                                      float g3, float g4) {
    float r = 0.0f;
    r = (d == 0) ? g0 : r;
    r = (d == 1) ? g1 : r;
    r = (d == 2) ? g2 : r;
    r = (d == 3) ? g3 : r;
    r = (d == 4) ? g4 : r;
    return r;
}

__launch_bounds__(128, 1)
__global__ void gauss3d_sep_wmma(const float* __restrict__ xin,
                                 const float* __restrict__ wx,
                                 const float* __restrict__ wy,
                                 const float* __restrict__ wz,
                                 float* __restrict__ out)
{
    __shared__ float gtap[8];                 // 1D taps (5 used)
    __shared__ float bufA[HS * HS * HS];      // 8000 f32: halo input, later stage-Y output
    __shared__ float bufB[HS * HS * TS];      // 6400 f32: stage-X output, later stage-Z output

    const int tid  = threadIdx.x;
    const int lane = tid & 31;
    // Wave id as an SGPR so the per-wave chain loops are scalar branches
    // (EXEC never modified around the WMMAs, as the ISA requires).
    const int wv   = __builtin_amdgcn_readfirstlane(tid) >> 5;   // 0..3
    const int m    = lane & 15;    // row/col index within 16
    const int sub  = lane >> 4;    // lane half -> K sub-block {0,1} vs {2,3}

    const int id = blockIdx.x;
    const int ch = id >> 12;                       // channel 0..2
    const int t3 = id & 4095;
    const int tz = t3 >> 8, ty = (t3 >> 4) & 15, tx = t3 & 15;
    const int z0 = tz * TS, y0 = ty * TS, x0 = tx * TS;

    const float* w   = (ch == 0) ? wx : (ch == 1) ? wy : wz;
    const float* src = xin + (size_t)ch * ((size_t)DIMSZ * DIMSZ * DIMSZ);

    if (tid == 0) __builtin_prefetch(w, 0, 0);     // global_prefetch_b8

    // Recover the exact 1D taps: g[i] = sum_{j,k} w3[i][j][k]  (w3 sums to 1)
    if (tid < NTAP) {
        float s = 0.0f;
        #pragma unroll
        for (int q = 0; q < 25; ++q) s += w[tid * 25 + q];
        gtap[tid] = s;
    }

    // Load 20x20x20 haloed input tile with zero padding outside the volume.
    for (int i = tid; i < HS * HS * HS; i += 128) {
        int zz = i / (HS * HS);
        int rr = i - zz * (HS * HS);
        int yy = rr / HS;
        int xx = rr - yy * HS;
        int gz = z0 + zz - 2, gy = y0 + yy - 2, gx = x0 + xx - 2;
        float v = 0.0f;
        if ((unsigned)gz < DIMSZ && (unsigned)gy < DIMSZ && (unsigned)gx < DIMSZ)
            v = src[((size_t)gz << 16) + (gy << 8) + gx];
        bufA[i] = v;
    }
    __syncthreads();

    // Taps into registers, then build all five B fragments ONCE in registers.
    // B layout (4x16, 2 VGPRs): lanes<16 hold K-rows {4b, 4b+1}, lanes>=16
    // hold K-rows {4b+2, 4b+3}; column = m. G[t,j] = g[t-j] on the band.
    const float g0 = gtap[0], g1 = gtap[1], g2 = gtap[2],
                g3 = gtap[3], g4 = gtap[4];
    v2f bfrag[5];
    #pragma unroll
    for (int b = 0; b < 5; ++b) {
        const int t0 = 4 * b + 2 * sub;
        bfrag[b].x = pick(t0 - m,     g0, g1, g2, g3, g4);
        bfrag[b].y = pick(t0 + 1 - m, g0, g1, g2, g3, g4);
    }

    // ---- Stage X: conv along x. 20x20 (z,y) lines -> 25 chains of 16 lines.
    // bufB[z][y][j] = sum_d g[d] * bufA[z][y][j+d]
    for (int cc = wv; cc < 25; cc += 4) {
        const int l  = cc * 16 + m;
        const int zi = l / HS, yi = l - zi * HS;
        const int ab = zi * (HS * HS) + yi * HS;
        v8f acc = {};
        #pragma unroll
        for (int b = 0; b < 5; ++b) {
            const int t0 = 4 * b + 2 * sub;
            v2f a;  a.x = bufA[ab + t0];  a.y = bufA[ab + t0 + 1];
            acc = __builtin_amdgcn_wmma_f32_16x16x4_f32(
                      false, a, false, bfrag[b], (short)0, acc, false, false);
        }
        #pragma unroll
        for (int r = 0; r < 8; ++r) {
            const int l2 = cc * 16 + r + 8 * sub;
            const int z2 = l2 / HS, y2 = l2 - z2 * HS;
            bufB[z2 * (HS * TS) + y2 * TS + m] = acc[r];
        }
    }
    __syncthreads();

    // ---- Stage Y: conv along y. 20x16 (z,x) lines -> 20 chains.
    // bufA[z][j][x] = sum_d g[d] * bufB[z][j+d][x]   (bufA reused, 20*16*16)
    for (int cc = wv; cc < 20; cc += 4) {
        const int l  = cc * 16 + m;
        const int zi = l >> 4, xi = l & 15;
        const int ab = zi * (HS * TS) + xi;
        v8f acc = {};
        #pragma unroll
        for (int b = 0; b < 5; ++b) {
            const int t0 = 4 * b + 2 * sub;
            v2f a;  a.x = bufB[ab + t0 * TS];  a.y = bufB[ab + (t0 + 1) * TS];
            acc = __builtin_amdgcn_wmma_f32_16x16x4_f32(
                      false, a, false, bfrag[b], (short)0, acc, false, false);
        }
        #pragma unroll
        for (int r = 0; r < 8; ++r) {
            const int l2 = cc * 16 + r + 8 * sub;
            const int z2 = l2 >> 4, x2 = l2 & 15;
            bufA[z2 * 256 + m * 16 + x2] = acc[r];
        }
    }
    __syncthreads();

    // ---- Stage Z: conv along z. 16x16 (y,x) lines -> 16 chains.
    // bufB[j][y][x] = sum_d g[d] * bufA[j+d][y][x]   (bufB reused, 16^3)
    for (int cc = wv; cc < 16; cc += 4) {
        const int l  = cc * 16 + m;
        const int yi = l >> 4, xi = l & 15;
        const int ab = yi * 16 + xi;
        v8f acc = {};
        #pragma unroll
        for (int b = 0; b < 5; ++b) {
            const int t0 = 4 * b + 2 * sub;
            v2f a;  a.x = bufA[t0 * 256 + ab];  a.y = bufA[(t0 + 1) * 256 + ab];
            acc = __builtin_amdgcn_wmma_f32_16x16x4_f32(
                      false, a, false, bfrag[b], (short)0, acc, false, false);
        }
        #pragma unroll
        for (int r = 0; r < 8; ++r) {
            const int l2 = cc * 16 + r + 8 * sub;
            const int y2 = l2 >> 4, x2 = l2 & 15;
            bufB[m * 256 + y2 * 16 + x2] = acc[r];
        }
    }
    __syncthreads();

    // ---- Coalesced float4 store of the 16^3 tile.
    const float4* b4 = (const float4*)bufB;
    float* dst = out + (size_t)ch * ((size_t)DIMSZ * DIMSZ * DIMSZ);
    for (int i = tid; i < 1024; i += 128) {
        const int zz = i >> 6;
        const int yy = (i >> 2) & 15;
        const int xq = (i & 3) << 2;
        *(float4*)(dst + ((size_t)(z0 + zz) << 16) + ((y0 + yy) << 8) + (x0 + xq)) = b4[i];
    }
}

extern "C" void kernel_launch(void* const* d_in, const int* in_sizes, int n_in,
                              void* d_out, int out_size, void* d_ws, size_t ws_size,
                              hipStream_t stream) {
    const float* x  = (const float*)d_in[0];
    const float* wx = (const float*)d_in[1];
    const float* wy = (const float*)d_in[2];
    const float* wz = (const float*)d_in[3];
    float* out = (float*)d_out;

    const int tilesPerChan = 16 * 16 * 16;           // 256/16 per dim
    dim3 grid(3 * tilesPerChan);                     // 12288 blocks
    dim3 block(128);                                 // 4 waves (wave32)
    gauss3d_sep_wmma<<<grid, block, 0, stream>>>(x, wx, wy, wz, out);
}